// VNGNN_25520695673457
// MI455X (gfx1250) — compile-verified
//
#include <hip/hip_runtime.h>
#include <hip/hip_bf16.h>

#define FEAT 128
#define NV   8
#define EPS  1e-5f
#define RPB  256          // rows per GEMM block
#define SAPAD 132         // padded LDS row stride (floats) -> conflict-free b64 frag loads

typedef __attribute__((ext_vector_type(2))) float v2f;
typedef __attribute__((ext_vector_type(8))) float v8f;

// ---------------------------------------------------------------------------
// degree / dinv
// ---------------------------------------------------------------------------
__global__ void deg_init_kernel(float* deg, int n) {
    int i = blockIdx.x * blockDim.x + threadIdx.x;
    if (i < n) deg[i] = 1.0f;                      // self-loop contributes 1
}

__global__ void deg_accum_kernel(const int* __restrict__ dst, float* deg, int e) {
    int i = blockIdx.x * blockDim.x + threadIdx.x;
    if (i < e) atomicAdd(&deg[dst[i]], 1.0f);
}

__global__ void dinv_kernel(float* deg, int n) {
    int i = blockIdx.x * blockDim.x + threadIdx.x;
    if (i < n) deg[i] = rsqrtf(deg[i]);            // deg >= 1 always (self-loops)
}

// ---------------------------------------------------------------------------
// small helpers
// ---------------------------------------------------------------------------
__global__ void zero_kernel(float* p, int n) {
    int i = blockIdx.x * blockDim.x + threadIdx.x;
    if (i < n) p[i] = 0.0f;
}

__global__ void copy_kernel(float* dst, const float* __restrict__ src, int n) {
    int i = blockIdx.x * blockDim.x + threadIdx.x;
    if (i < n) dst[i] = src[i];
}

__global__ void vnsum_kernel(float* vnsum, const float* __restrict__ vn) {
    int c = threadIdx.x;                            // 128 threads
    float s = 0.f;
#pragma unroll
    for (int v = 0; v < NV; ++v) s += vn[v * FEAT + c];
    vnsum[c] = s;
}

// h[i,c] = emb[i,c] + vnsum[c]
__global__ __launch_bounds__(FEAT) void addvn_kernel(float* __restrict__ h,
                                                     const float* __restrict__ emb,
                                                     const float* __restrict__ vnsum,
                                                     int n) {
    int i = blockIdx.x;
    int c = threadIdx.x;
    h[(size_t)i * FEAT + c] = emb[(size_t)i * FEAT + c] + vnsum[c];
}

// ---------------------------------------------------------------------------
// GEMM: hl[N x 128] = h[N x 128] @ W[128 x 128] via V_WMMA_F32_16X16X4_F32.
// Block = 256 threads = 8 waves. Wave w owns column tile n in [16w, 16w+16):
// its B fragments (64 VGPRs) are preloaded once from L2-resident W. The block
// then loops over 16-row strips of A, staging each strip coalesced into
// padded LDS; per k-step the A fragment is one conflict-free ds_load_b64.
// Inner loop: 32 x (ds_load_b64 + v_wmma_f32_16x16x4_f32) per wave per strip.
// ---------------------------------------------------------------------------
__global__ __launch_bounds__(256) void gemm_wmma_kernel(const float* __restrict__ h,
                                                        const float* __restrict__ W,
                                                        float* __restrict__ hl,
                                                        int nrows) {
    __shared__ float sA[16 * SAPAD];                // 16 rows, padded stride

    const int tid  = threadIdx.x;
    const int wave = tid >> 5;                      // column tile 0..7
    const int lane = tid & 31;
    const int lm   = lane & 15;
    const int hi   = lane >> 4;                     // 0/1
    const int klo  = 2 * hi;
    const int n    = wave * 16 + lm;                // output column of this lane

    // --- preload B column tile into registers (one-time, W is L2-resident) ---
    v2f barr[32];
#pragma unroll
    for (int i = 0; i < 32; ++i) {
        int k = 4 * i + klo;
        v2f b;
        b.x = W[(size_t)k * FEAT + n];
        b.y = W[(size_t)(k + 1) * FEAT + n];
        barr[i] = b;
    }

    const int base = blockIdx.x * RPB;
    int nvalid = nrows - base;
    if (nvalid > RPB) nvalid = RPB;
    const int nstrips = (nvalid + 15) >> 4;         // uniform across the block

    for (int s = 0; s < nstrips; ++s) {
        const int row0 = base + s * 16;

        __syncthreads();                            // sA safe to overwrite
        // stage A strip: 16 x 128 floats, coalesced float4, zero-pad tail rows
        for (int i = tid; i < 16 * 32; i += 256) {
            int r  = i >> 5;                        // strip row 0..15
            int c4 = i & 31;                        // float4 column
            int row = row0 + r;
            float4 v;
            if (row < nrows) v = ((const float4*)(h + (size_t)row * FEAT))[c4];
            else { v.x = 0.f; v.y = 0.f; v.z = 0.f; v.w = 0.f; }
            *(float4*)&sA[r * SAPAD + c4 * 4] = v;
        }
        __syncthreads();

        // prefetch next strip into cache while we compute (global_prefetch_b8)
        int nrow0 = row0 + 16;
        if (nrow0 < nrows)
            __builtin_prefetch(h + (size_t)nrow0 * FEAT + tid * 8, 0, 1);

        v8f acc = {0.f, 0.f, 0.f, 0.f, 0.f, 0.f, 0.f, 0.f};
#pragma unroll
        for (int kp = 0; kp < 32; ++kp) {
            int k = 4 * kp + klo;
            v2f a = *(const v2f*)&sA[lm * SAPAD + k];   // ds_load_b64, conflict-free
            acc = __builtin_amdgcn_wmma_f32_16x16x4_f32(
                false, a, false, barr[kp], (short)0, acc, false, false);
        }

        // C/D layout: acc[r] = C[row0 + r + 8*hi][n]
        const int mb = row0 + 8 * hi;
#pragma unroll
        for (int r = 0; r < 8; ++r) {
            int rr = mb + r;
            if (rr < nrows) hl[(size_t)rr * FEAT + n] = acc[r];
        }
    }
}

// ---------------------------------------------------------------------------
// scatter: agg = bc + self-loop + sum_edges  (sym-normalized)
// ---------------------------------------------------------------------------
__global__ __launch_bounds__(FEAT) void scatter_init_kernel(const float* __restrict__ hl,
                                                            const float* __restrict__ dinv,
                                                            const float* __restrict__ bias,
                                                            float* __restrict__ agg,
                                                            int n) {
    int i = blockIdx.x;
    int c = threadIdx.x;
    float dn = dinv[i];
    agg[(size_t)i * FEAT + c] = bias[c] + hl[(size_t)i * FEAT + c] * dn * dn;
}

// one wave per edge; each lane handles 4 channels (float4)
__global__ __launch_bounds__(256) void scatter_edges_kernel(const int* __restrict__ src,
                                                            const int* __restrict__ dst,
                                                            const float* __restrict__ dinv,
                                                            const float* __restrict__ hl,
                                                            float* __restrict__ agg,
                                                            int e) {
    int edge = blockIdx.x * 8 + (threadIdx.x >> 5);
    if (edge >= e) return;
    int lane = threadIdx.x & 31;
    int s = src[edge];
    int d = dst[edge];
    float nrm = dinv[s] * dinv[d];
    float4 v = ((const float4*)(hl + (size_t)s * FEAT))[lane];
    float* base = agg + (size_t)d * FEAT + lane * 4;
    atomicAdd(base + 0, v.x * nrm);
    atomicAdd(base + 1, v.y * nrm);
    atomicAdd(base + 2, v.z * nrm);
    atomicAdd(base + 3, v.w * nrm);
}

// ---------------------------------------------------------------------------
// batch norm (training-mode batch stats over nodes)
// ---------------------------------------------------------------------------
__global__ __launch_bounds__(FEAT) void bn_stats_kernel(const float* __restrict__ agg,
                                                        float* __restrict__ sums,   // [2*F]
                                                        int n) {
    int c = threadIdx.x;
    float s = 0.f, ss = 0.f;
    for (int i = blockIdx.x; i < n; i += gridDim.x) {
        float v = agg[(size_t)i * FEAT + c];
        s  += v;
        ss += v * v;
    }
    atomicAdd(&sums[c], s);
    atomicAdd(&sums[FEAT + c], ss);
}

__global__ void bn_finalize_kernel(const float* __restrict__ sums,
                                   const float* __restrict__ g,
                                   const float* __restrict__ b,
                                   float* __restrict__ scsh,   // [2*F]: scale, shift
                                   int n) {
    int c = threadIdx.x;
    float inv_n = 1.0f / (float)n;
    float mean = sums[c] * inv_n;
    float var  = sums[FEAT + c] * inv_n - mean * mean;
    float scale = g[c] * rsqrtf(var + EPS);
    scsh[c]        = scale;
    scsh[FEAT + c] = b[c] - mean * scale;
}

__global__ __launch_bounds__(FEAT) void bn_apply_kernel(const float* __restrict__ agg,
                                                        const float* __restrict__ scsh,
                                                        float* __restrict__ out,
                                                        int n, int do_relu) {
    int i = blockIdx.x;
    int c = threadIdx.x;
    float v = agg[(size_t)i * FEAT + c] * scsh[c] + scsh[FEAT + c];
    if (do_relu) v = fmaxf(v, 0.0f);
    out[(size_t)i * FEAT + c] = v;
}

// ---------------------------------------------------------------------------
// masked global-add pool: pooled[v,c] = sum_i mask[v,i] * emb[i,c]
// (jnp bool array -> 1 byte per element)
// ---------------------------------------------------------------------------
__global__ __launch_bounds__(FEAT) void pool_kernel(const unsigned char* __restrict__ mask,
                                                    const float* __restrict__ emb,
                                                    float* __restrict__ pooled,
                                                    int n) {
    int c = threadIdx.x;
    float acc[NV];
#pragma unroll
    for (int v = 0; v < NV; ++v) acc[v] = 0.f;
    for (int i = blockIdx.x; i < n; i += gridDim.x) {
        float val = emb[(size_t)i * FEAT + c];
#pragma unroll
        for (int v = 0; v < NV; ++v)
            if (mask[(size_t)v * n + i]) acc[v] += val;
    }
#pragma unroll
    for (int v = 0; v < NV; ++v) atomicAdd(&pooled[v * FEAT + c], acc[v]);
}

// ---------------------------------------------------------------------------
// per-VN MLP: vn = LN(relu(LN(relu((pooled+vn)@W1+b1))@W2+b2))
// one block per virtual node (8 blocks x 256 threads)
// ---------------------------------------------------------------------------
__device__ __forceinline__ float blk_sum256(float* red, float v, int t) {
    red[t] = v;
    __syncthreads();
    for (int s = 128; s >= 1; s >>= 1) {
        if (t < s) red[t] += red[t + s];
        __syncthreads();
    }
    float r = red[0];
    __syncthreads();
    return r;
}

__global__ __launch_bounds__(256) void vn_mlp_kernel(const float* __restrict__ pooled,
                                                     float* __restrict__ vn,
                                                     const float* __restrict__ W1,
                                                     const float* __restrict__ b1,
                                                     const float* __restrict__ g1,
                                                     const float* __restrict__ bb1,
                                                     const float* __restrict__ W2,
                                                     const float* __restrict__ b2,
                                                     const float* __restrict__ g2,
                                                     const float* __restrict__ bb2) {
    const int H = 2 * FEAT;                         // 256
    int v = blockIdx.x;                             // 0..7
    int t = threadIdx.x;                            // 0..255

    __shared__ float s_tmp[FEAT];
    __shared__ float s_h1[2 * FEAT];
    __shared__ float red[256];

    if (t < FEAT) s_tmp[t] = pooled[v * FEAT + t] + vn[v * FEAT + t];
    __syncthreads();

    // h1[t] = relu(sum_f tmp[f] * W1[v][f][t] + b1[v][t]),  t in [0,256)
    float a = b1[v * H + t];
    const float* w1 = W1 + (size_t)v * FEAT * H + t;
#pragma unroll 4
    for (int f = 0; f < FEAT; ++f) a += s_tmp[f] * w1[(size_t)f * H];
    a = fmaxf(a, 0.0f);

    // LayerNorm over 256
    float sum = blk_sum256(red, a, t);
    float sq  = blk_sum256(red, a * a, t);
    float m1  = sum * (1.0f / H);
    float var1 = sq * (1.0f / H) - m1 * m1;
    s_h1[t] = (a - m1) * rsqrtf(var1 + EPS) * g1[v * H + t] + bb1[v * H + t];
    __syncthreads();

    // h2[t] = relu(sum_j h1[j] * W2[v][j][t] + b2[v][t]),  t in [0,128)
    float o = 0.0f;
    if (t < FEAT) {
        o = b2[v * FEAT + t];
        const float* w2 = W2 + (size_t)v * H * FEAT + t;
#pragma unroll 4
        for (int j = 0; j < H; ++j) o += s_h1[j] * w2[(size_t)j * FEAT];
        o = fmaxf(o, 0.0f);
    }
    // LayerNorm over 128 (lanes >= 128 contribute 0)
    float sum2 = blk_sum256(red, o, t);
    float sq2  = blk_sum256(red, o * o, t);
    float m2   = sum2 * (1.0f / FEAT);
    float var2 = sq2 * (1.0f / FEAT) - m2 * m2;
    if (t < FEAT)
        vn[v * FEAT + t] = (o - m2) * rsqrtf(var2 + EPS) * g2[v * FEAT + t] + bb2[v * FEAT + t];
}

// ---------------------------------------------------------------------------
// host orchestration
// ---------------------------------------------------------------------------
extern "C" void kernel_launch(void* const* d_in, const int* in_sizes, int n_in,
                              void* d_out, int out_size, void* d_ws, size_t ws_size,
                              hipStream_t stream) {
    const float*         x    = (const float*)d_in[0];
    const int*           ei   = (const int*)d_in[1];
    const unsigned char* mask = (const unsigned char*)d_in[2];
    const float*         Wc   = (const float*)d_in[3];
    const float*         bc   = (const float*)d_in[4];
    const float*         bng  = (const float*)d_in[5];
    const float*         bnb  = (const float*)d_in[6];
    const float*         vne  = (const float*)d_in[7];
    const float*         W1   = (const float*)d_in[8];
    const float*         b1   = (const float*)d_in[9];
    const float*         g1   = (const float*)d_in[10];
    const float*         bb1  = (const float*)d_in[11];
    const float*         W2   = (const float*)d_in[12];
    const float*         b2   = (const float*)d_in[13];
    const float*         g2   = (const float*)d_in[14];   // ln2_g
    const float*         bb2  = (const float*)d_in[15];   // ln2_b

    const int N = in_sizes[0] / FEAT;
    const int E = in_sizes[1] / 2;
    const int* src = ei;
    const int* dst = ei + E;

    // workspace layout (floats)
    float* ws     = (float*)d_ws;
    float* dinv   = ws;                                   // N
    float* vn     = dinv + N;                             // V*F = 1024
    float* vnsum  = vn + NV * FEAT;                       // 128
    float* pooled = vnsum + FEAT;                         // 1024
    float* bnsums = pooled + NV * FEAT;                   // 256
    float* scsh   = bnsums + 2 * FEAT;                    // 256
    float* bufE   = scsh + 2 * FEAT;                      // N*F (emb ping)
    float* bufH   = bufE + (size_t)N * FEAT;              // N*F (h / agg)
    float* bufL   = bufH + (size_t)N * FEAT;              // N*F (hl)

    // degree -> dinv
    deg_init_kernel<<<(N + 255) / 256, 256, 0, stream>>>(dinv, N);
    deg_accum_kernel<<<(E + 255) / 256, 256, 0, stream>>>(dst, dinv, E);
    dinv_kernel<<<(N + 255) / 256, 256, 0, stream>>>(dinv, N);

    // vn = vn_embed
    copy_kernel<<<(NV * FEAT + 255) / 256, 256, 0, stream>>>(vn, vne, NV * FEAT);

    const float* emb_in = x;
    for (int l = 0; l < 3; ++l) {
        const int last = (l == 2);

        vnsum_kernel<<<1, FEAT, 0, stream>>>(vnsum, vn);
        addvn_kernel<<<N, FEAT, 0, stream>>>(bufH, emb_in, vnsum, N);

        gemm_wmma_kernel<<<(N + RPB - 1) / RPB, 256, 0, stream>>>(
            bufH, Wc + (size_t)l * FEAT * FEAT, bufL, N);

        if (!last) {
            zero_kernel<<<(NV * FEAT + 255) / 256, 256, 0, stream>>>(pooled, NV * FEAT);
            pool_kernel<<<512, FEAT, 0, stream>>>(mask, emb_in, pooled, N);
        }

        // agg (reuses bufH; h is dead after the GEMM)
        scatter_init_kernel<<<N, FEAT, 0, stream>>>(bufL, dinv, bc + l * FEAT, bufH, N);
        scatter_edges_kernel<<<(E + 7) / 8, 256, 0, stream>>>(src, dst, dinv, bufL, bufH, E);

        zero_kernel<<<1, 256, 0, stream>>>(bnsums, 2 * FEAT);
        bn_stats_kernel<<<512, FEAT, 0, stream>>>(bufH, bnsums, N);
        bn_finalize_kernel<<<1, FEAT, 0, stream>>>(bnsums, bng + l * FEAT, bnb + l * FEAT, scsh, N);

        float* emb_out = last ? (float*)d_out : bufE;
        bn_apply_kernel<<<N, FEAT, 0, stream>>>(bufH, scsh, emb_out, N, !last);

        if (!last) {
            const int H = 2 * FEAT;
            vn_mlp_kernel<<<NV, 256, 0, stream>>>(
                pooled, vn,
                W1 + (size_t)l * NV * FEAT * H, b1 + (size_t)l * NV * H,
                g1 + (size_t)l * NV * H,        bb1 + (size_t)l * NV * H,
                W2 + (size_t)l * NV * H * FEAT, b2 + (size_t)l * NV * FEAT,
                g2 + (size_t)l * NV * FEAT,     bb2 + (size_t)l * NV * FEAT);
        }
        emb_in = bufE;
    }
}